// AttentiveGraphPooling_80564996538938
// MI455X (gfx1250) — compile-verified
//
#include <hip/hip_runtime.h>
#include <hip/hip_bf16.h>

#define N_GRAPH 256
#define NVERT   8192
#define NEMB    128
#define WAVES   8            // waves per block in pool kernel (256 threads)
#define VPW     (NVERT / WAVES)

typedef __attribute__((ext_vector_type(2))) float v2f;
typedef __attribute__((ext_vector_type(4))) float v4f;
typedef __attribute__((ext_vector_type(8))) float v8f;

__device__ __forceinline__ float fast_tanh(float x) {
    // tanh(x) = 1 - 2/(exp(2x)+1); exp via v_exp_f32 (exp2), inverse via v_rcp_f32.
    // Saturates correctly for large |x| (exp2->inf/0), NaN propagates.
    float e = __builtin_amdgcn_exp2f(x * 2.885390081777927f); // 2*log2(e)
    return 1.0f - 2.0f * __builtin_amdgcn_rcpf(e + 1.0f);
}

__device__ __forceinline__ float fast_exp(float x) {
    return __builtin_amdgcn_exp2f(x * 1.4426950408889634f);
}

// ---------------------------------------------------------------------------
// Kernel 1: fused additive-tanh attention + online softmax + weighted pooling.
// One block per graph n. 8 waves; each wave owns a contiguous 1024-vertex
// chunk and keeps a private online-softmax state (m, s, acc[lane*4..+3]).
// Wave partials are merged through LDS; leaky_relu'd heads -> workspace.
// ---------------------------------------------------------------------------
__global__ __launch_bounds__(256) void pool_kernel(
    const float* __restrict__ vertices, const float* __restrict__ query,
    const float* __restrict__ W_logit, const float* __restrict__ b_logit,
    const float* __restrict__ temp, float* __restrict__ heads)
{
    const int n    = blockIdx.x;
    const int tid  = threadIdx.x;
    const int lane = tid & 31;
    const int wave = tid >> 5;
    const int e0   = lane * 4;

    __shared__ float s_m[WAVES];
    __shared__ float s_s[WAVES];
    __shared__ __align__(16) float s_acc[WAVES][NEMB];

    const v4f q  = *(const v4f*)(query   + e0);
    const v4f wl = *(const v4f*)(W_logit + e0);
    const float bl   = b_logit[0];
    const float invT = __builtin_amdgcn_rcpf(temp[0]);

    const v4f* vp = (const v4f*)(vertices
                      + (size_t)n * NVERT * NEMB
                      + (size_t)(wave * VPW) * NEMB + e0);

    float m = -__builtin_inff();
    float ssum = 0.0f;
    v4f acc = {0.0f, 0.0f, 0.0f, 0.0f};

    for (int v = 0; v < VPW; ++v) {
        // streaming load: 16B/lane, coalesced 512B/row/wave, non-temporal
        v4f x = __builtin_nontemporal_load(vp);
        vp += NEMB / 4;

        float r;
        r  = fast_tanh(x.x + q.x) * wl.x;
        r += fast_tanh(x.y + q.y) * wl.y;
        r += fast_tanh(x.z + q.z) * wl.z;
        r += fast_tanh(x.w + q.w) * wl.w;

        // wave32 reduction over the 128-element dot product
        r += __shfl_xor(r, 16, 32);
        r += __shfl_xor(r, 8, 32);
        r += __shfl_xor(r, 4, 32);
        r += __shfl_xor(r, 2, 32);
        r += __shfl_xor(r, 1, 32);

        const float z = (r + bl) * invT;

        // online softmax update (first iter: m=-inf -> c=0, state reset to p)
        const float mnew = fmaxf(m, z);
        const float c = fast_exp(m - mnew);
        const float p = fast_exp(z - mnew);
        ssum = ssum * c + p;
        acc.x = acc.x * c + p * x.x;
        acc.y = acc.y * c + p * x.y;
        acc.z = acc.z * c + p * x.z;
        acc.w = acc.w * c + p * x.w;
        m = mnew;
    }

    if (lane == 0) { s_m[wave] = m; s_s[wave] = ssum; }
    *(v4f*)(&s_acc[wave][e0]) = acc;
    __syncthreads();

    // merge the 8 wave-local softmax states; threads 0..127 own one e each
    if (tid < NEMB) {
        float M = s_m[0];
        #pragma unroll
        for (int w = 1; w < WAVES; ++w) M = fmaxf(M, s_m[w]);
        float S = 0.0f, pooled = 0.0f;
        #pragma unroll
        for (int w = 0; w < WAVES; ++w) {
            const float c = fast_exp(s_m[w] - M);
            S      += s_s[w] * c;
            pooled += s_acc[w][tid] * c;
        }
        pooled *= __builtin_amdgcn_rcpf(S);
        const float h = pooled > 0.0f ? pooled : 0.01f * pooled;  // leaky_relu
        heads[n * NEMB + tid] = h;
    }
}

// ---------------------------------------------------------------------------
// Kernel 2: out[n,j] = sum_k heads[n,k] * W_reduce[j,k] + b_reduce[j]
// V_WMMA_F32_16X16X4_F32, one wave per 16x16 output tile, K-loop of 32 WMMAs.
// A 16x4 f32 layout: lanes 0-15 -> M=lane, VGPR{0,1}=K{0,1};
//                    lanes 16-31 -> M=lane-16, VGPR{0,1}=K{2,3}.
// B 4x16 mirrored (rows striped across lanes, low K in lanes 0-15).
// C/D: 8 VGPRs, N = lane&15, M = r + 8*(lane>>4).
// ---------------------------------------------------------------------------
__global__ __launch_bounds__(32) void reduce_gemm_kernel(
    const float* __restrict__ heads, const float* __restrict__ W_reduce,
    const float* __restrict__ b_reduce, float* __restrict__ out)
{
    const int lane = threadIdx.x;          // full wave32, EXEC all ones
    const int tile = blockIdx.x;           // 0..127
    const int j0 = (tile & 7) * 16;        // N tile over output features
    const int n0 = (tile >> 3) * 16;       // M tile over graphs
    const int half = lane >> 4;
    const int l    = lane & 15;

    const float* Arow = heads    + (size_t)(n0 + l) * NEMB;  // A[m=l][k]
    const float* Brow = W_reduce + (size_t)(j0 + l) * NEMB;  // B[k][j=l] = W[j][k]

    v8f c = {0.0f, 0.0f, 0.0f, 0.0f, 0.0f, 0.0f, 0.0f, 0.0f};

    #pragma unroll 4
    for (int kk = 0; kk < NEMB / 4; ++kk) {
        const int k = kk * 4 + half * 2;
        v2f a; a.x = Arow[k]; a.y = Arow[k + 1];
        v2f b; b.x = Brow[k]; b.y = Brow[k + 1];
        c = __builtin_amdgcn_wmma_f32_16x16x4_f32(
                /*neg_a=*/false, a, /*neg_b=*/false, b,
                /*c_mod=*/(short)0, c, /*reuse_a=*/false, /*reuse_b=*/false);
    }

    const float bias = b_reduce[j0 + l];
    #pragma unroll
    for (int r = 0; r < 8; ++r) {
        const int mm = r + 8 * half;
        out[(size_t)(n0 + mm) * NEMB + j0 + l] = c[r] + bias;
    }
}

extern "C" void kernel_launch(void* const* d_in, const int* in_sizes, int n_in,
                              void* d_out, int out_size, void* d_ws, size_t ws_size,
                              hipStream_t stream) {
    const float* vertices = (const float*)d_in[0];
    const float* query    = (const float*)d_in[1];
    const float* W_logit  = (const float*)d_in[2];
    const float* b_logit  = (const float*)d_in[3];
    const float* W_reduce = (const float*)d_in[4];
    const float* b_reduce = (const float*)d_in[5];
    const float* temp     = (const float*)d_in[6];

    float* out   = (float*)d_out;
    float* heads = (float*)d_ws;   // 256*128 floats = 128 KB scratch

    pool_kernel<<<N_GRAPH, 256, 0, stream>>>(vertices, query, W_logit, b_logit,
                                             temp, heads);
    reduce_gemm_kernel<<<(N_GRAPH / 16) * (NEMB / 16), 32, 0, stream>>>(
        heads, W_reduce, b_reduce, out);
}